// GNNMotionPredictor_15676630630712
// MI455X (gfx1250) — compile-verified
//
#include <hip/hip_runtime.h>

// ---------------------------------------------------------------------------
// GINEConv GNN on gfx1250 (MI455X), f32 WMMA path, persistent blocks.
//
//   e      = relu(a * We1 + be1) @ (We2 @ Wl) + (be2 @ Wl + bl)   [per edge]
//   m      = relu(x[src] + e)         -> atomicAdd into haccum[dst]
//   h      = x + aggr   (eps = 0)     -> haccum initialized with x (DMA copy)
//   h      = relu(h @ Wn1 + bn1) @ Wn2 + bn2
//   out    = h @ Wo + bo              -> [N, 3]
//
// GEMMs use V_WMMA_F32_16X16X4_F32: M=16 rows/wave, K=64 (16 steps of 4),
// N=64 (4 accumulator tiles). Weights live in LDS (stride-65 padded),
// loaded once per persistent block.
// ---------------------------------------------------------------------------

typedef float v2f __attribute__((ext_vector_type(2)));
typedef float v8f __attribute__((ext_vector_type(8)));

__device__ __forceinline__ v8f wmma4(v2f a, v2f b, v8f c) {
  // (neg_a, A, neg_b, B, c_mod, C, reuse_a, reuse_b)
  return __builtin_amdgcn_wmma_f32_16x16x4_f32(false, a, false, b, (short)0, c,
                                               false, false);
}

__device__ __forceinline__ v8f bcast8(float x) {
  v8f r;
#pragma unroll
  for (int i = 0; i < 8; ++i) r[i] = x;
  return r;
}

// ---------------------------------------------------------------------------
// Kernel 0: fuse edge-encoder tail + GINE lin:  W2l = We2 @ Wl,
//           b2l = be2 @ Wl + bl.   (single block, tiny)
// ---------------------------------------------------------------------------
__global__ void __launch_bounds__(256) prep_weights(
    const float* __restrict__ We2, const float* __restrict__ Wl,
    const float* __restrict__ be2, const float* __restrict__ bl,
    float* __restrict__ W2l, float* __restrict__ b2l) {
  int t = threadIdx.x;
  for (int idx = t; idx < 64 * 64; idx += 256) {
    int j = idx >> 6, k = idx & 63;
    float s = 0.f;
#pragma unroll 8
    for (int i = 0; i < 64; ++i) s = fmaf(We2[j * 64 + i], Wl[i * 64 + k], s);
    W2l[idx] = s;
  }
  if (t < 64) {
    float s = bl[t];
#pragma unroll 8
    for (int i = 0; i < 64; ++i) s = fmaf(be2[i], Wl[i * 64 + t], s);
    b2l[t] = s;
  }
}

// ---------------------------------------------------------------------------
// Kernel 1: edge pipeline. 256 threads = 8 waves, 16 edges per wave-batch,
// persistent grid-stride over batches.
// ---------------------------------------------------------------------------
__global__ void __launch_bounds__(256) edge_kernel(
    const float* __restrict__ x, const long long* __restrict__ ei,
    const float* __restrict__ ea, const float* __restrict__ We1,
    const float* __restrict__ be1, const float* __restrict__ W2l,
    const float* __restrict__ b2l, float* __restrict__ haccum, int E) {
  __shared__ float sWe1[64];
  __shared__ float sbe1[64];
  __shared__ float sb2l[64];
  __shared__ float sW2l[64 * 65];  // stride 65: conflict-free B reads

  int t = threadIdx.x;
  if (t < 64) {
    sWe1[t] = We1[t];
    sbe1[t] = be1[t];
    sb2l[t] = b2l[t];
  }
  for (int i = t; i < 64 * 64; i += 256)
    sW2l[(i >> 6) * 65 + (i & 63)] = W2l[i];
  __syncthreads();

  int wave = t >> 5, lane = t & 31;
  int m16 = lane & 15;  // row (edge-in-batch) / column-in-tile for this lane
  int hi = lane >> 4;   // K-pair selector
  long long nBatches = ((long long)E + 15) >> 4;

  for (long long batch = (long long)blockIdx.x * 8 + wave; batch < nBatches;
       batch += (long long)gridDim.x * 8) {
    long long base = batch * 16;
    long long e = base + m16;
    long long ec = (e < E) ? e : (long long)(E - 1);
    float a = ea[ec];
    int srcn = (int)ei[ec];
    int dstn = (int)ei[(long long)E + ec];

    // Warm the x[src] rows we will gather after the GEMM (global_prefetch_b8).
    __builtin_prefetch(x + (long long)srcn * 64, 0, 3);
    __builtin_prefetch(x + (long long)srcn * 64 + 32, 0, 3);

    v8f acc[4];
#pragma unroll
    for (int nt = 0; nt < 4; ++nt) acc[nt] = bcast8(sb2l[nt * 16 + m16]);

#pragma unroll
    for (int kk = 0; kk < 16; ++kk) {
      int c0 = kk * 4 + hi * 2;
      // A element [m16][c0], [m16][c0+1] computed in-register: relu(a*w + b)
      v2f A;
      A.x = fmaxf(fmaf(a, sWe1[c0], sbe1[c0]), 0.f);
      A.y = fmaxf(fmaf(a, sWe1[c0 + 1], sbe1[c0 + 1]), 0.f);
#pragma unroll
      for (int nt = 0; nt < 4; ++nt) {
        v2f B;
        B.x = sW2l[c0 * 65 + nt * 16 + m16];
        B.y = sW2l[(c0 + 1) * 65 + nt * 16 + m16];
        acc[nt] = wmma4(A, B, acc[nt]);
      }
    }

    // Scatter: m = relu(x[src] + e); atomicAdd into haccum[dst].
    // D layout: VGPR r, lanes<16 -> row r, col=lane; lanes>=16 -> row r+8.
#pragma unroll
    for (int r = 0; r < 8; ++r) {
      int m = r + hi * 8;
      int sm = __shfl(srcn, m, 32);
      int dm = __shfl(dstn, m, 32);
      int vm = (base + (long long)m) < (long long)E;
#pragma unroll
      for (int nt = 0; nt < 4; ++nt) {
        int n = nt * 16 + m16;
        float msg = fmaxf(x[(long long)sm * 64 + n] + acc[nt][r], 0.f);
        if (vm) atomicAdd(&haccum[(long long)dm * 64 + n], msg);
      }
    }
  }
}

// ---------------------------------------------------------------------------
// Kernel 2: node MLP + head. 128 threads = 4 waves, 16 nodes per wave,
// persistent grid-stride over 64-node block-batches.
// ---------------------------------------------------------------------------
__global__ void __launch_bounds__(128) node_kernel(
    const float* __restrict__ h, const float* __restrict__ Wn1,
    const float* __restrict__ bn1, const float* __restrict__ Wn2,
    const float* __restrict__ bn2, const float* __restrict__ Wo,
    const float* __restrict__ bo, float* __restrict__ out, int N) {
  __shared__ float sW1[64 * 65];
  __shared__ float sW2[64 * 65];
  __shared__ float sb1[64];
  __shared__ float sb2[64];
  __shared__ float sWo[192];
  __shared__ float sbo[3];
  __shared__ float st[4][16 * 65];  // per-wave 16x64 tile, stride 65

  int t = threadIdx.x;
  for (int i = t; i < 64 * 64; i += 128) {
    sW1[(i >> 6) * 65 + (i & 63)] = Wn1[i];
    sW2[(i >> 6) * 65 + (i & 63)] = Wn2[i];
  }
  if (t < 64) {
    sb1[t] = bn1[t];
    sb2[t] = bn2[t];
  }
  if (t < 3) sbo[t] = bo[t];
  for (int i = t; i < 192; i += 128) sWo[i] = Wo[i];
  __syncthreads();

  int wave = t >> 5, lane = t & 31;
  int m16 = lane & 15;
  int hi = lane >> 4;
  long long nBatches = ((long long)N + 63) >> 6;  // 64 nodes per block-batch

  for (long long batch = blockIdx.x; batch < nBatches;
       batch += (long long)gridDim.x) {
    long long base = batch * 64 + (long long)wave * 16;
    long long row = base + m16;
    long long rowC = (row < N) ? row : (long long)(N - 1);
    const float2* hrow = (const float2*)(h + rowC * 64);

    // ---- GEMM 1: t = relu(h @ Wn1 + bn1) ----
    v8f acc[4];
#pragma unroll
    for (int nt = 0; nt < 4; ++nt) acc[nt] = bcast8(sb1[nt * 16 + m16]);
#pragma unroll
    for (int kk = 0; kk < 16; ++kk) {
      int c0 = kk * 4 + hi * 2;
      float2 hp = hrow[c0 >> 1];
      v2f A;
      A.x = hp.x;
      A.y = hp.y;
#pragma unroll
      for (int nt = 0; nt < 4; ++nt) {
        v2f B;
        B.x = sW1[c0 * 65 + nt * 16 + m16];
        B.y = sW1[(c0 + 1) * 65 + nt * 16 + m16];
        acc[nt] = wmma4(A, B, acc[nt]);
      }
    }
    // relu, spill D-layout -> LDS so it can be reloaded in A-layout.
    // st[wave] is touched only by this wave; same-wave LDS ops are in-order,
    // so a compiler scheduling barrier is sufficient (no block barrier).
#pragma unroll
    for (int r = 0; r < 8; ++r) {
      int m = r + hi * 8;
#pragma unroll
      for (int nt = 0; nt < 4; ++nt)
        st[wave][m * 65 + nt * 16 + m16] = fmaxf(acc[nt][r], 0.f);
    }
    __builtin_amdgcn_wave_barrier();

    // ---- GEMM 2: h2 = t @ Wn2 + bn2 ----
#pragma unroll
    for (int nt = 0; nt < 4; ++nt) acc[nt] = bcast8(sb2[nt * 16 + m16]);
#pragma unroll
    for (int kk = 0; kk < 16; ++kk) {
      int c0 = kk * 4 + hi * 2;
      v2f A;
      A.x = st[wave][m16 * 65 + c0];
      A.y = st[wave][m16 * 65 + c0 + 1];
#pragma unroll
      for (int nt = 0; nt < 4; ++nt) {
        v2f B;
        B.x = sW2[c0 * 65 + nt * 16 + m16];
        B.y = sW2[(c0 + 1) * 65 + nt * 16 + m16];
        acc[nt] = wmma4(A, B, acc[nt]);
      }
    }
    __builtin_amdgcn_wave_barrier();
    // spill h2 (in-order LDS: the A reads above complete before these writes)
#pragma unroll
    for (int r = 0; r < 8; ++r) {
      int m = r + hi * 8;
#pragma unroll
      for (int nt = 0; nt < 4; ++nt)
        st[wave][m * 65 + nt * 16 + m16] = acc[nt][r];
    }
    __builtin_amdgcn_wave_barrier();

    // ---- Output head: out[node, j] = h2 . Wo[:, j] + bo[j] ----
    for (int item = lane; item < 48; item += 32) {
      int mrow = item / 3, j = item - 3 * (item / 3);
      long long node = base + mrow;
      if (node < N) {
        float s = sbo[j];
#pragma unroll 8
        for (int c = 0; c < 64; ++c)
          s = fmaf(st[wave][mrow * 65 + c], sWo[c * 3 + j], s);
        out[node * 3 + j] = s;
      }
    }
    __builtin_amdgcn_wave_barrier();  // keep next iter's writes behind reads
  }
}

// ---------------------------------------------------------------------------
extern "C" void kernel_launch(void* const* d_in, const int* in_sizes, int n_in,
                              void* d_out, int out_size, void* d_ws,
                              size_t ws_size, hipStream_t stream) {
  const float* x = (const float*)d_in[0];
  const long long* ei = (const long long*)d_in[1];
  const float* ea = (const float*)d_in[2];
  const float* We2 = (const float*)d_in[5];
  const float* be2 = (const float*)d_in[6];
  const float* Wl = (const float*)d_in[7];
  const float* bl = (const float*)d_in[8];
  float* out = (float*)d_out;

  int N = in_sizes[0] / 64;
  int E = in_sizes[2];

  // Workspace layout: haccum[N*64] | W2l[64*64] | b2l[64]
  float* haccum = (float*)d_ws;
  float* W2l = haccum + (size_t)N * 64;
  float* b2l = W2l + 64 * 64;

  prep_weights<<<1, 256, 0, stream>>>(We2, Wl, be2, bl, W2l, b2l);

  // haccum = x  (eps == 0 -> h = x + aggr); DMA copy engine path.
  hipMemcpyAsync(haccum, x, (size_t)N * 64 * sizeof(float),
                 hipMemcpyDeviceToDevice, stream);

  // Persistent edge kernel: 8 wave-batches of 16 edges per block pass.
  long long eBatches = ((long long)E + 15) >> 4;
  int eBlocksNeeded = (int)((eBatches + 7) >> 3);
  int eGrid = eBlocksNeeded < 1024 ? eBlocksNeeded : 1024;
  edge_kernel<<<eGrid, 256, 0, stream>>>(
      x, ei, ea, (const float*)d_in[3], (const float*)d_in[4], W2l, b2l,
      haccum, E);

  // Persistent node kernel: 64 nodes per block-batch.
  long long nBatches = ((long long)N + 63) >> 6;
  int nGrid = nBatches < 512 ? (int)nBatches : 512;
  node_kernel<<<nGrid, 128, 0, stream>>>(
      haccum, (const float*)d_in[9], (const float*)d_in[10],
      (const float*)d_in[11], (const float*)d_in[12], (const float*)d_in[13],
      (const float*)d_in[14], out, N);
}